// DifferentiableSuperpixelTokenizer_34557306863963
// MI455X (gfx1250) — compile-verified
//
#include <hip/hip_runtime.h>

#define NSEG   196
#define NB     8
#define IMH    224
#define IMW    224
#define NPIX   (IMH * IMW)       // 50176
#define EMB    768
#define ROWS   (NB * NSEG)       // 1568  (= 98 * 16 exactly)
#define CHUNKS 32
#define PIX_PER_CHUNK (NPIX / CHUNKS)  // 1568

typedef __attribute__((ext_vector_type(2))) float v2f;
typedef __attribute__((ext_vector_type(8))) float v8f;

// ---------------------------------------------------------------------------
// Kernel 0: zero the per-(batch,segment) accumulator scratch.
// ws layout: float ws[ROWS][8] = {sum_c0, sum_c1, sum_c2, sum_x, sum_y, count, 0, 0}
// ---------------------------------------------------------------------------
__global__ void zero_ws_kernel(float* __restrict__ ws, int n) {
    int i = blockIdx.x * blockDim.x + threadIdx.x;
    if (i < n) ws[i] = 0.0f;
}

// ---------------------------------------------------------------------------
// Kernel 1: segment-sum of 5-dim pixel features + pixel count.
// grid = NB * CHUNKS blocks, 256 threads. LDS accumulation (ds_add_f32),
// then one global float atomic per non-zero LDS cell.
// ---------------------------------------------------------------------------
__global__ __launch_bounds__(256)
void seg_accum_kernel(const float* __restrict__ img,
                      const int*   __restrict__ seg,
                      float*       __restrict__ ws) {
    __shared__ float acc[NSEG * 6];
    const int t = threadIdx.x;
    for (int i = t; i < NSEG * 6; i += 256) acc[i] = 0.0f;
    __syncthreads();

    const int b     = blockIdx.x / CHUNKS;
    const int chunk = blockIdx.x % CHUNKS;
    const int p0    = chunk * PIX_PER_CHUNK;

    const float* imgb = img + (size_t)b * 3 * NPIX;   // [C, H*W] planes
    const int*   segb = seg + (size_t)b * NPIX;

    const float invw = 1.0f / (float)(IMW - 1);
    const float invh = 1.0f / (float)(IMH - 1);

    for (int p = p0 + t; p < p0 + PIX_PER_CHUNK; p += 256) {
        const int   s  = segb[p];
        const float r  = imgb[p];
        const float g  = imgb[NPIX + p];
        const float bl = imgb[2 * NPIX + p];
        const float x  = (float)(p % IMW) * invw;
        const float y  = (float)(p / IMW) * invh;
        atomicAdd(&acc[s * 6 + 0], r);
        atomicAdd(&acc[s * 6 + 1], g);
        atomicAdd(&acc[s * 6 + 2], bl);
        atomicAdd(&acc[s * 6 + 3], x);
        atomicAdd(&acc[s * 6 + 4], y);
        atomicAdd(&acc[s * 6 + 5], 1.0f);
    }
    __syncthreads();

    for (int i = t; i < NSEG * 6; i += 256) {
        const int   s = i / 6;
        const int   j = i % 6;
        const float v = acc[i];
        if (v != 0.0f)
            atomicAdd(&ws[(size_t)(b * NSEG + s) * 8 + j], v);
    }
}

// ---------------------------------------------------------------------------
// Kernel 2: [1568, 5(pad 8)] @ [5, 768] + masked bias, via two chained
// V_WMMA_F32_16X16X4_F32 per 16x16 output tile. One wave32 per tile.
// grid = (48, 98), block = 32 (EXEC all-ones as WMMA requires).
// A-fragment layout (16x4 f32): lanes 0-15 -> K={0,1}, lanes 16-31 -> K={2,3}.
// B symmetric; C/D: VGPR j holds M=j (lanes 0-15) and M=j+8 (lanes 16-31).
// ---------------------------------------------------------------------------
__global__ __launch_bounds__(32)
void seg_gemm_wmma_kernel(const float* __restrict__ ws,
                          const float* __restrict__ Wm,    // [5, 768] row-major
                          const float* __restrict__ bias,  // [768]
                          float*       __restrict__ out) { // [1568, 768]
    const int lane   = threadIdx.x;          // 0..31
    const int tile_n = blockIdx.x;           // 0..47
    const int tile_m = blockIdx.y;           // 0..97

    const int mrow = tile_m * 16 + (lane & 15);
    const int ncol = tile_n * 16 + (lane & 15);
    const int kk   = (lane >> 4) * 2;        // 0 (lanes 0-15) or 2 (lanes 16-31)

    const float* wsm = ws + (size_t)mrow * 8;
    const float  cnt = wsm[5];
    const float  inv = 1.0f / fmaxf(cnt, 1.0f);

    // A fragments: K block 0..3 and K block 4..7 (K>=5 padded with zeros)
    v2f a0, a1, b0, b1;
    a0.x = wsm[kk + 0] * inv;
    a0.y = wsm[kk + 1] * inv;
    a1.x = (kk == 0) ? wsm[4] * inv : 0.0f;  // K=4 real, K=6 pad
    a1.y = 0.0f;                             // K=5 / K=7 pad

    // B fragments from W (row-major [K, 768])
    b0.x = Wm[(kk + 0) * EMB + ncol];
    b0.y = Wm[(kk + 1) * EMB + ncol];
    b1.x = (kk == 0) ? Wm[4 * EMB + ncol] : 0.0f;
    b1.y = 0.0f;

    v8f c = {};
    c = __builtin_amdgcn_wmma_f32_16x16x4_f32(false, a0, false, b0,
                                              (short)0, c, false, false);
    c = __builtin_amdgcn_wmma_f32_16x16x4_f32(false, a1, false, b1,
                                              (short)0, c, false, false);

    const float bv    = bias[ncol];
    const int   mbase = tile_m * 16 + ((lane >> 4) ? 8 : 0);
#pragma unroll
    for (int j = 0; j < 8; ++j) {
        const int   m  = mbase + j;
        const float cj = ws[(size_t)m * 8 + 5];
        out[(size_t)m * EMB + ncol] = (cj > 0.0f) ? (c[j] + bv) : 0.0f;
    }
}

// ---------------------------------------------------------------------------
extern "C" void kernel_launch(void* const* d_in, const int* in_sizes, int n_in,
                              void* d_out, int out_size, void* d_ws, size_t ws_size,
                              hipStream_t stream) {
    (void)in_sizes; (void)n_in; (void)out_size; (void)ws_size;
    const float* img  = (const float*)d_in[0];   // [8,3,224,224] f32
    const int*   seg  = (const int*)  d_in[1];   // [8,224,224]   i32
    const float* Wm   = (const float*)d_in[2];   // [5,768]       f32
    const float* bias = (const float*)d_in[3];   // [768]         f32
    float*       out  = (float*)d_out;           // [8,196,768]   f32
    float*       ws   = (float*)d_ws;            // >= 1568*8 floats

    const int ws_elems = ROWS * 8;
    zero_ws_kernel<<<(ws_elems + 255) / 256, 256, 0, stream>>>(ws, ws_elems);

    seg_accum_kernel<<<NB * CHUNKS, 256, 0, stream>>>(img, seg, ws);

    dim3 grid(EMB / 16, ROWS / 16);   // (48, 98)
    seg_gemm_wmma_kernel<<<grid, 32, 0, stream>>>(ws, Wm, bias, out);
}